// DiyMaxPooling_5909875000105
// MI455X (gfx1250) — compile-verified
//
#include <hip/hip_runtime.h>

// 2x2 stride-2 NCHW max pooling, fp32.
// Input  (16, 64, 512, 512)  -> 1 GiB
// Output (16, 64, 256, 256)  -> 256 MiB
// Pure streaming (zero reuse): optimized for 23.3 TB/s HBM with b128
// non-temporal loads/stores, wave32-contiguous addressing.

typedef float v4f __attribute__((ext_vector_type(4)));

namespace {
constexpr unsigned N  = 16;
constexpr unsigned C  = 64;
constexpr unsigned H  = 512;
constexpr unsigned W  = 512;
constexpr unsigned OH = H / 2;
constexpr unsigned OW = W / 2;
constexpr unsigned OWQ = OW / 4;                    // float4 groups per output row
constexpr unsigned TOTAL_OUT_VEC = N * C * OH * OWQ; // 16,777,216 threads
}

__global__ __launch_bounds__(256) void
maxpool2x2_nt_kernel(const float* __restrict__ in, float* __restrict__ out) {
    unsigned idx = blockIdx.x * blockDim.x + threadIdx.x;
    if (idx >= TOTAL_OUT_VEC) return;

    // idx -> (plane = n*C+c, oh, group-of-4 ow)
    unsigned owq   = idx % OWQ;          // 0..63
    unsigned r     = idx / OWQ;          // output row id across all planes
    unsigned oh    = r % OH;
    unsigned plane = r / OH;             // 0..N*C-1

    // Input rows 2*oh and 2*oh+1, starting at x = owq*8 (32B aligned).
    const float* row0 = in + ((size_t)plane * H + (size_t)oh * 2u) * W + (size_t)owq * 8u;
    const float* row1 = row0 + W;

    // Four 128-bit non-temporal loads (streaming: data never re-read, input
    // exceeds the 192MB L2 -> keep it out of the cache hierarchy).
    v4f a0 = __builtin_nontemporal_load((const v4f*)(row0));
    v4f a1 = __builtin_nontemporal_load((const v4f*)(row0 + 4));
    v4f b0 = __builtin_nontemporal_load((const v4f*)(row1));
    v4f b1 = __builtin_nontemporal_load((const v4f*)(row1 + 4));

    // Vertical max then horizontal max of adjacent pairs -> 4 outputs.
    v4f t0, t1;
    t0.x = fmaxf(a0.x, b0.x);
    t0.y = fmaxf(a0.y, b0.y);
    t0.z = fmaxf(a0.z, b0.z);
    t0.w = fmaxf(a0.w, b0.w);
    t1.x = fmaxf(a1.x, b1.x);
    t1.y = fmaxf(a1.y, b1.y);
    t1.z = fmaxf(a1.z, b1.z);
    t1.w = fmaxf(a1.w, b1.w);

    v4f o;
    o.x = fmaxf(t0.x, t0.y);
    o.y = fmaxf(t0.z, t0.w);
    o.z = fmaxf(t1.x, t1.y);
    o.w = fmaxf(t1.z, t1.w);

    // One 128-bit non-temporal store; wave covers 512 B contiguous output.
    __builtin_nontemporal_store(o, (v4f*)(out + (size_t)idx * 4u));
}

extern "C" void kernel_launch(void* const* d_in, const int* in_sizes, int n_in,
                              void* d_out, int out_size, void* d_ws, size_t ws_size,
                              hipStream_t stream) {
    (void)in_sizes; (void)n_in; (void)d_ws; (void)ws_size;
    const float* in = (const float*)d_in[0];
    float* out = (float*)d_out;

    // out_size = 67,108,864 elements; 4 per thread -> 16,777,216 threads.
    unsigned total_vec = (unsigned)(out_size / 4);
    unsigned block = 256;                          // 8 wave32 per block
    unsigned grid  = (total_vec + block - 1) / block;
    maxpool2x2_nt_kernel<<<grid, block, 0, stream>>>(in, out);
}